// GAT_87771951661273
// MI455X (gfx1250) — compile-verified
//
#include <hip/hip_runtime.h>
#include <math.h>

typedef __attribute__((ext_vector_type(2))) float v2f;
typedef __attribute__((ext_vector_type(8))) float v8f;

#define NEG_SLOPE 0.2f
#define GAT_EPS 1e-16f

// ---------------------------------------------------------------- utilities

__global__ void k_fill(float* __restrict__ p, float v, int n) {
    int i = blockIdx.x * blockDim.x + threadIdx.x;
    if (i < n) p[i] = v;
}

// src/dst for edge e; self-loops appended after the E real edges
__device__ __forceinline__ void edge_sd(const int* __restrict__ ei, int E, int e,
                                        int& s, int& d) {
    if (e < E) { s = ei[e]; d = ei[E + e]; }
    else       { s = e - E; d = e - E; }
}

// order-exact float atomic max via monotone integer bit trick (init = -inf)
__device__ __forceinline__ void atomicMaxF(float* addr, float val) {
    if (val >= 0.0f)
        atomicMax((int*)addr, __float_as_int(val));
    else
        atomicMin((unsigned int*)addr, __float_as_uint(val));
}

// ---------------------------------------------------------------- GEMM 1
// h[N,16] = x[N,128] @ W[128,16], exact fp32 via V_WMMA_F32_16X16X4_F32.
// One wave per 16-row tile; 32 WMMA steps (K=4 each).
// A layout (ISA 7.12.2, 32-bit A 16x4): lanes 0-15 -> K=0(v0),K=1(v1);
// lanes 16-31 -> K=2(v0),K=3(v1). B mirrored. C: VGPR r = row r (lanes 0-15)
// / row r+8 (lanes 16-31), col = lane%16.
__global__ void k_gemm1(const float* __restrict__ x, const float* __restrict__ W,
                        float* __restrict__ h, int n) {
    int lane = threadIdx.x & 31;
    int wave = threadIdx.x >> 5;
    int tile = blockIdx.x * (blockDim.x >> 5) + wave;
    if (tile * 16 >= n) return;
    int half = lane >> 4;
    int lm   = lane & 15;
    int row  = tile * 16 + lm; if (row >= n) row = n - 1; // rows independent: clamp ok
    const float* xr = x + (size_t)row * 128;

    v8f c = {};
#pragma unroll 8
    for (int k = 0; k < 128; k += 4) {
        int k0 = k + half * 2;
        v2f a, b;
        a.x = xr[k0];
        a.y = xr[k0 + 1];
        b.x = W[(size_t)k0 * 16 + lm];
        b.y = W[(size_t)(k0 + 1) * 16 + lm];
        c = __builtin_amdgcn_wmma_f32_16x16x4_f32(false, a, false, b,
                                                  (short)0, c, false, false);
    }
    // store epilogue: wave-uniform fast path (no per-element exec juggling)
    if (tile * 16 + 16 <= n) {
        float* hb = h + ((size_t)(tile * 16 + half * 8)) * 16 + lm;
#pragma unroll
        for (int r = 0; r < 8; ++r) hb[(size_t)r * 16] = c[r];
    } else {
#pragma unroll
        for (int r = 0; r < 8; ++r) {
            int orow = tile * 16 + r + half * 8;
            if (orow < n) h[(size_t)orow * 16 + lm] = c[r];
        }
    }
}

// ---------------------------------------------------------------- GEMM 2
// o[N,128] = h[N,16] @ W[16,128]. blockDim=256: wave w handles column tile
// n0 = w*16 of row tile blockIdx.x. K=16 -> 4 WMMA steps.
__global__ void k_gemm2(const float* __restrict__ h, const float* __restrict__ W,
                        float* __restrict__ o, int n) {
    int lane = threadIdx.x & 31;
    int wave = threadIdx.x >> 5;      // column tile 0..7
    int tile = blockIdx.x;            // row tile
    if (tile * 16 >= n) return;
    int half = lane >> 4;
    int lm   = lane & 15;
    int row  = tile * 16 + lm; if (row >= n) row = n - 1;
    const float* hr = h + (size_t)row * 16;
    int n0 = wave * 16;

    v8f c = {};
#pragma unroll
    for (int k = 0; k < 16; k += 4) {
        int k0 = k + half * 2;
        v2f a, b;
        a.x = hr[k0];
        a.y = hr[k0 + 1];
        b.x = W[(size_t)k0 * 128 + n0 + lm];
        b.y = W[(size_t)(k0 + 1) * 128 + n0 + lm];
        c = __builtin_amdgcn_wmma_f32_16x16x4_f32(false, a, false, b,
                                                  (short)0, c, false, false);
    }
    if (tile * 16 + 16 <= n) {
        float* ob = o + ((size_t)(tile * 16 + half * 8)) * 128 + n0 + lm;
#pragma unroll
        for (int r = 0; r < 8; ++r) ob[(size_t)r * 128] = c[r];
    } else {
#pragma unroll
        for (int r = 0; r < 8; ++r) {
            int orow = tile * 16 + r + half * 8;
            if (orow < n) o[(size_t)orow * 128 + n0 + lm] = c[r];
        }
    }
}

// ------------------------------------------------- per-node attention dots
// s_src[v] = h[v,:].a_src ; s_dst[v] = h[v,:].a_dst. One wave32 per node.
__global__ void k_dots(const float* __restrict__ h, const float* __restrict__ asr,
                       const float* __restrict__ adst, float* __restrict__ ssrc,
                       float* __restrict__ sdst, int n, int F) {
    int lane = threadIdx.x & 31;
    int node = blockIdx.x * (blockDim.x >> 5) + (threadIdx.x >> 5);
    if (node >= n) return;
    const float* hr = h + (size_t)node * F;
    float ps = 0.0f, pd = 0.0f;
    for (int f = lane; f < F; f += 32) {
        float v = hr[f];
        ps += v * asr[f];
        pd += v * adst[f];
    }
    for (int o = 16; o > 0; o >>= 1) {
        ps += __shfl_down(ps, o, 32);
        pd += __shfl_down(pd, o, 32);
    }
    if (lane == 0) { ssrc[node] = ps; sdst[node] = pd; }
}

// ------------------------------------------------------------- edge passes
__global__ void k_edge_alpha_max(const int* __restrict__ ei, int E, int M,
                                 const float* __restrict__ ssrc,
                                 const float* __restrict__ sdst,
                                 float* __restrict__ alpha, float* __restrict__ mseg) {
    int e = blockIdx.x * blockDim.x + threadIdx.x;
    if (e >= M) return;
    int s, d; edge_sd(ei, E, e, s, d);
    float a = ssrc[s] + sdst[d];
    a = (a > 0.0f) ? a : a * NEG_SLOPE;   // leaky_relu
    alpha[e] = a;
    atomicMaxF(mseg + d, a);
}

__global__ void k_edge_exp_sum(const int* __restrict__ ei, int E, int M,
                               float* __restrict__ alpha,
                               const float* __restrict__ mseg,
                               float* __restrict__ ssum) {
    int e = blockIdx.x * blockDim.x + threadIdx.x;
    if (e >= M) return;
    int s, d; edge_sd(ei, E, e, s, d);
    float v = expf(alpha[e] - mseg[d]);
    alpha[e] = v;                          // store exp value in place
    atomicAdd(ssum + d, v);
}

// layer-1 aggregation: 16 lanes per edge (F=16), L2-resident gather+scatter
__global__ void k_edge_agg16(const int* __restrict__ ei, int E, int M,
                             const float* __restrict__ alpha,
                             const float* __restrict__ ssum,
                             const float* __restrict__ h, float* __restrict__ agg) {
    int t = blockIdx.x * blockDim.x + threadIdx.x;
    int e = t >> 4, f = t & 15;
    if (e >= M) return;
    int s, d; edge_sd(ei, E, e, s, d);
    float w = alpha[e] / (ssum[d] + GAT_EPS);
    atomicAdd(agg + (size_t)d * 16 + f, h[(size_t)s * 16 + f] * w);
}

// layer-2 aggregation: one wave per edge, float4 per lane (16B/lane vectorized)
__global__ void k_edge_agg128(const int* __restrict__ ei, int E, int M,
                              const float* __restrict__ alpha,
                              const float* __restrict__ ssum,
                              const float* __restrict__ h, float* __restrict__ outp) {
    int lane = threadIdx.x & 31;
    int e = blockIdx.x * (blockDim.x >> 5) + (threadIdx.x >> 5);
    if (e >= M) return;
    int s, d; edge_sd(ei, E, e, s, d);
    float w = alpha[e] / (ssum[d] + GAT_EPS);
    const float4* hs = (const float4*)(h + (size_t)s * 128);
    float4 v = hs[lane];                              // global_load_b128
    float* o = outp + (size_t)d * 128 + (size_t)lane * 4;
    atomicAdd(o + 0, v.x * w);
    atomicAdd(o + 1, v.y * w);
    atomicAdd(o + 2, v.z * w);
    atomicAdd(o + 3, v.w * w);
}

// ---------------------------------------------------------------- epilogues
__global__ void k_bias_relu(float* __restrict__ a, const float* __restrict__ b,
                            int n, int fmask) {
    int i = blockIdx.x * blockDim.x + threadIdx.x;
    if (i < n) {
        float v = a[i] + b[i & fmask];
        a[i] = v > 0.0f ? v : 0.0f;
    }
}

__global__ void k_bias(float* __restrict__ a, const float* __restrict__ b,
                       int n, int fmask) {
    int i = blockIdx.x * blockDim.x + threadIdx.x;
    if (i < n) a[i] += b[i & fmask];
}

// ---------------------------------------------------------------- driver

extern "C" void kernel_launch(void* const* d_in, const int* in_sizes, int n_in,
                              void* d_out, int out_size, void* d_ws, size_t ws_size,
                              hipStream_t stream) {
    const float* x   = (const float*)d_in[0];
    const int*   ei  = (const int*)  d_in[1];
    const float* W1  = (const float*)d_in[2];
    const float* as1 = (const float*)d_in[3];
    const float* ad1 = (const float*)d_in[4];
    const float* b1  = (const float*)d_in[5];
    const float* W2  = (const float*)d_in[6];
    const float* as2 = (const float*)d_in[7];
    const float* ad2 = (const float*)d_in[8];
    const float* b2  = (const float*)d_in[9];
    float* outp = (float*)d_out;

    const int N = in_sizes[0] / 128;   // 100000
    const int E = in_sizes[1] / 2;     // 3200000
    const int M = E + N;               // + self loops

    // workspace carve-up (~79 MB total)
    float* ws = (float*)d_ws;
    size_t off = 0;
    float* h1    = ws + off; off += (size_t)N * 16;
    float* agg1  = ws + off; off += (size_t)N * 16;   // relu'd -> layer-2 input
    float* h2    = ws + off; off += (size_t)N * 128;
    float* ssrc  = ws + off; off += (size_t)N;
    float* sdst  = ws + off; off += (size_t)N;
    float* mseg  = ws + off; off += (size_t)N;
    float* ssum  = ws + off; off += (size_t)N;
    float* alpha = ws + off; off += (size_t)M;
    (void)ws_size; (void)n_in; (void)out_size;

    const int B = 256;
    const int tilesN = (N + 15) / 16;
    const int gN   = (N + B - 1) / B;
    const int gM   = (M + B - 1) / B;

    // -------------------- layer 1 --------------------
    k_gemm1<<<dim3((tilesN + 7) / 8), B, 0, stream>>>(x, W1, h1, N);
    k_dots <<<dim3((N + 7) / 8),      B, 0, stream>>>(h1, as1, ad1, ssrc, sdst, N, 16);
    k_fill <<<dim3(gN), B, 0, stream>>>(mseg, -INFINITY, N);
    hipMemsetAsync(ssum, 0, (size_t)N * 4, stream);
    hipMemsetAsync(agg1, 0, (size_t)N * 16 * 4, stream);
    k_edge_alpha_max<<<dim3(gM), B, 0, stream>>>(ei, E, M, ssrc, sdst, alpha, mseg);
    k_edge_exp_sum  <<<dim3(gM), B, 0, stream>>>(ei, E, M, alpha, mseg, ssum);
    k_edge_agg16    <<<dim3((M * 16 + B - 1) / B), B, 0, stream>>>(ei, E, M, alpha, ssum, h1, agg1);
    k_bias_relu     <<<dim3((N * 16 + B - 1) / B), B, 0, stream>>>(agg1, b1, N * 16, 15);

    // -------------------- layer 2 --------------------
    k_gemm2<<<dim3(tilesN),      B, 0, stream>>>(agg1, W2, h2, N);
    k_dots <<<dim3((N + 7) / 8), B, 0, stream>>>(h2, as2, ad2, ssrc, sdst, N, 128);
    k_fill <<<dim3(gN), B, 0, stream>>>(mseg, -INFINITY, N);
    hipMemsetAsync(ssum, 0, (size_t)N * 4, stream);
    hipMemsetAsync(d_out, 0, (size_t)N * 128 * 4, stream);
    k_edge_alpha_max<<<dim3(gM), B, 0, stream>>>(ei, E, M, ssrc, sdst, alpha, mseg);
    k_edge_exp_sum  <<<dim3(gM), B, 0, stream>>>(ei, E, M, alpha, mseg, ssum);
    k_edge_agg128   <<<dim3((M + 7) / 8), B, 0, stream>>>(ei, E, M, alpha, ssum, h2, outp);
    k_bias          <<<dim3(((size_t)N * 128 + B - 1) / B), B, 0, stream>>>(outp, b2, N * 128, 127);
}